// GNN_Classifier_20289425506743
// MI455X (gfx1250) — compile-verified
//
#include <hip/hip_runtime.h>
#include <hip/hip_bf16.h>

#define N_NODES_   50000
#define N_GRAPHS_  64
#define DIM_       128

typedef float v2f __attribute__((ext_vector_type(2)));
typedef float v8f __attribute__((ext_vector_type(8)));

// ---------------------------------------------------------------- utilities
__global__ void fill_kernel(float* __restrict__ p, float v, int n) {
    int i = blockIdx.x * blockDim.x + threadIdx.x;
    if (i < n) p[i] = v;
}

__global__ void copy4_kernel(const float4* __restrict__ s, float4* __restrict__ d, int n4) {
    int i = blockIdx.x * blockDim.x + threadIdx.x;
    if (i < n4) d[i] = s[i];
}

__global__ void invert_kernel(float* __restrict__ c, int n) {
    int i = blockIdx.x * blockDim.x + threadIdx.x;
    if (i < n) c[i] = 1.0f / fmaxf(c[i], 1.0f);
}

// --------------------------------------------------------- degree counting
__global__ void count_kernel(const int* __restrict__ dst, float* __restrict__ cnt, int n_edges) {
    int i = blockIdx.x * blockDim.x + threadIdx.x;
    if (i < n_edges) unsafeAtomicAdd(&cnt[dst[i]], 1.0f);
}

// ------------------------------------------------- edge scatter (wave/edge)
// agg[dst[e]][:] += x[src[e]][:]  (agg pre-initialized with x for self-loops)
__global__ void __launch_bounds__(256)
scatter_kernel(const float* __restrict__ x, const int* __restrict__ src,
               const int* __restrict__ dst, float* __restrict__ agg, int n_edges) {
    int wid  = (blockIdx.x * blockDim.x + threadIdx.x) >> 5;
    int lane = threadIdx.x & 31;
    if (wid >= n_edges) return;
    int s = src[wid], d = dst[wid];
    const float4 v = ((const float4*)(x + (size_t)s * DIM_))[lane];  // 512B/wave contiguous
    float* p = agg + (size_t)d * DIM_ + lane * 4;
    unsafeAtomicAdd(p + 0, v.x);
    unsafeAtomicAdd(p + 1, v.y);
    unsafeAtomicAdd(p + 2, v.z);
    unsafeAtomicAdd(p + 3, v.w);
}

// ------------------------------------- fused mean-scale + GEMM + bias + ReLU
// out[n][o] = relu( sum_k (agg[n][k]*invc[n]) * W[o][k] + b[o] )
// One block = 16 node rows x 128 out cols; wave w -> cols [16w,16w+16).
// V_WMMA_F32_16X16X4_F32, k-loop unrolled (32 steps).
__global__ void __launch_bounds__(256)
gemm_relu_kernel(const float* __restrict__ agg, const float* __restrict__ invc,
                 const float* __restrict__ W, const float* __restrict__ bias,
                 float* __restrict__ out, int n_nodes) {
    const int wave = threadIdx.x >> 5;      // 0..7
    const int lane = threadIdx.x & 31;
    const int half = lane >> 4;             // 0: K,K+1   1: K+2,K+3
    const int l15  = lane & 15;
    const int rowTile = blockIdx.x * 16;
    const int colBase = wave * 16;

    const int arow = min(rowTile + l15, n_nodes - 1);     // uniform EXEC for WMMA
    const float sc = invc[arow];
    const float* aptr = agg + (size_t)arow * DIM_ + 2 * half;
    const float* bptr = W + (size_t)(colBase + l15) * DIM_ + 2 * half;  // B[k][n] = W[n][k]

    v8f c = {};
#pragma unroll
    for (int k = 0; k < DIM_; k += 4) {
        v2f a, b;
        a.x = aptr[k] * sc;  a.y = aptr[k + 1] * sc;
        b.x = bptr[k];       b.y = bptr[k + 1];
        c = __builtin_amdgcn_wmma_f32_16x16x4_f32(false, a, false, b,
                                                  (short)0, c, false, false);
    }

    const int ocol = colBase + l15;
    const float bv = bias[ocol];
#pragma unroll
    for (int j = 0; j < 8; ++j) {
        int orow = rowTile + j + 8 * half;   // D vgpr j: row j (+8 for upper half-lanes)
        if (orow < n_nodes) {
            float v = c[j] + bv;
            out[(size_t)orow * DIM_ + ocol] = v > 0.0f ? v : 0.0f;
        }
    }
}

// ------------------------------------------------------- graph mean pooling
__global__ void __launch_bounds__(256)
pool_kernel(const float* __restrict__ h, const int* __restrict__ batch,
            float* __restrict__ gsum, float* __restrict__ gcnt, int n_nodes) {
    int wid  = (blockIdx.x * blockDim.x + threadIdx.x) >> 5;
    int lane = threadIdx.x & 31;
    if (wid >= n_nodes) return;
    int g = batch[wid];
    const float4 v = ((const float4*)(h + (size_t)wid * DIM_))[lane];
    float* p = gsum + (size_t)g * DIM_ + lane * 4;
    unsafeAtomicAdd(p + 0, v.x);
    unsafeAtomicAdd(p + 1, v.y);
    unsafeAtomicAdd(p + 2, v.z);
    unsafeAtomicAdd(p + 3, v.w);
    if (lane == 0) unsafeAtomicAdd(&gcnt[g], 1.0f);
}

// -------------------------------------------------------------- classifier
__global__ void __launch_bounds__(128)
head_kernel(const float* __restrict__ gsum, const float* __restrict__ gcnt,
            const float* __restrict__ Wc, const float* __restrict__ bc,
            float* __restrict__ out) {
    __shared__ float m[DIM_];
    const int g = blockIdx.x;
    const int t = threadIdx.x;
    const float inv = 1.0f / fmaxf(gcnt[g], 1.0f);
    float v = gsum[(size_t)g * DIM_ + t] * inv;
    m[t] = v > 0.0f ? v : 0.0f;
    __syncthreads();
    if (t < 10) {
        float acc = bc[t];
        const float* w = Wc + (size_t)t * DIM_;
#pragma unroll 8
        for (int k = 0; k < DIM_; ++k) acc += m[k] * w[k];
        out[g * 10 + t] = acc;
    }
}

// ------------------------------------------------------------------ launch
extern "C" void kernel_launch(void* const* d_in, const int* in_sizes, int n_in,
                              void* d_out, int out_size, void* d_ws, size_t ws_size,
                              hipStream_t stream) {
    const float* x    = (const float*)d_in[0];
    const int*   ei   = (const int*)d_in[1];
    const int*   batch= (const int*)d_in[2];
    const float* W1   = (const float*)d_in[3];
    const float* b1   = (const float*)d_in[4];
    const float* W2   = (const float*)d_in[5];
    const float* b2   = (const float*)d_in[6];
    const float* Wc   = (const float*)d_in[7];
    const float* bc   = (const float*)d_in[8];
    float* out = (float*)d_out;

    const int n_nodes = in_sizes[0] / DIM_;       // 50000
    const int n_edges = in_sizes[1] / 2;          // 600000
    const int* src = ei;
    const int* dst = ei + n_edges;

    // workspace carve-up (256B aligned)
    char* ws = (char*)d_ws;
    size_t featBytes = (size_t)n_nodes * DIM_ * sizeof(float);
    float* agg  = (float*)ws;                       ws += (featBytes + 255) & ~255ull;
    float* h    = (float*)ws;                       ws += (featBytes + 255) & ~255ull;
    float* cnt  = (float*)ws;                       ws += (((size_t)n_nodes * 4) + 255) & ~255ull;
    float* gsum = (float*)ws;                       ws += (N_GRAPHS_ * DIM_ * 4 + 255) & ~255ull;
    float* gcnt = (float*)ws;

    const int feat4    = n_nodes * (DIM_ / 4);      // float4 count
    const int scatterB = (n_edges * 32 + 255) / 256;
    const int poolB    = (n_nodes * 32 + 255) / 256;
    const int gemmB    = (n_nodes + 15) / 16;

    // degree counts (shared by both layers): cnt = 1 (self loop) + deg
    fill_kernel<<<(n_nodes + 255) / 256, 256, 0, stream>>>(cnt, 1.0f, n_nodes);
    fill_kernel<<<(N_GRAPHS_ * DIM_ + N_GRAPHS_ + 255) / 256, 256, 0, stream>>>(gsum, 0.0f,
                                                       N_GRAPHS_ * DIM_ + N_GRAPHS_);
    count_kernel<<<(n_edges + 255) / 256, 256, 0, stream>>>(dst, cnt, n_edges);
    invert_kernel<<<(n_nodes + 255) / 256, 256, 0, stream>>>(cnt, n_nodes);

    // layer 1
    copy4_kernel<<<(feat4 + 255) / 256, 256, 0, stream>>>((const float4*)x, (float4*)agg, feat4);
    scatter_kernel<<<scatterB, 256, 0, stream>>>(x, src, dst, agg, n_edges);
    gemm_relu_kernel<<<gemmB, 256, 0, stream>>>(agg, cnt, W1, b1, h, n_nodes);

    // layer 2 (h -> agg -> h, serialized on stream)
    copy4_kernel<<<(feat4 + 255) / 256, 256, 0, stream>>>((const float4*)h, (float4*)agg, feat4);
    scatter_kernel<<<scatterB, 256, 0, stream>>>(h, src, dst, agg, n_edges);
    gemm_relu_kernel<<<gemmB, 256, 0, stream>>>(agg, cnt, W2, b2, h, n_nodes);

    // pooling + head
    pool_kernel<<<poolB, 256, 0, stream>>>(h, batch, gsum, gcnt, n_nodes);
    head_kernel<<<N_GRAPHS_, 128, 0, stream>>>(gsum, gcnt, Wc, bc, out);
}